// GCN_27066883899968
// MI455X (gfx1250) — compile-verified
//
#include <hip/hip_runtime.h>

typedef __attribute__((ext_vector_type(2))) float v2f;
typedef __attribute__((ext_vector_type(8))) float v8f;

#define IN_FEATS 128
#define HID 32
#define WARPS_PER_BLOCK 4

// ---------------- utility kernels ----------------

__global__ void gcn_fill0(float* __restrict__ p, int n) {
    int t = blockIdx.x * blockDim.x + threadIdx.x;
    if (t < n) p[t] = 0.0f;
}

__global__ void gcn_degree(const int* __restrict__ src, const int* __restrict__ dst,
                           float* __restrict__ odeg, float* __restrict__ ideg, int E) {
    int e = blockIdx.x * blockDim.x + threadIdx.x;
    if (e >= E) return;
    atomicAdd(&odeg[src[e]], 1.0f);
    atomicAdd(&ideg[dst[e]], 1.0f);
}

__global__ void gcn_rsqrt(float* __restrict__ p, int n) {
    int t = blockIdx.x * blockDim.x + threadIdx.x;
    if (t < n) p[t] = rsqrtf(fmaxf(p[t], 1.0f));
}

// agg[dst] += h[src] * (use_scale ? out_norm[src] : 1)   (8 threads per edge, float4 each)
__global__ void gcn_spmm(const float* __restrict__ h, const int* __restrict__ src,
                         const int* __restrict__ dst, const float* __restrict__ onorm,
                         float* __restrict__ agg, int E, int use_scale) {
    int t = blockIdx.x * blockDim.x + threadIdx.x;
    int e = t >> 3;
    if (e >= E) return;
    int q = (t & 7) * 4;
    int s = src[e];
    int d = dst[e];
    const float4 v = *(const float4*)(h + s * HID + q);
    float sc = use_scale ? onorm[s] : 1.0f;
    float* base = agg + d * HID + q;
    atomicAdd(base + 0, v.x * sc);
    atomicAdd(base + 1, v.y * sc);
    atomicAdd(base + 2, v.z * sc);
    atomicAdd(base + 3, v.w * sc);
}

// h = agg * in_norm[node] + b1[feat]   (post conv1: weight already applied pre-agg)
__global__ void gcn_post1(const float* __restrict__ agg, const float* __restrict__ inorm,
                          const float* __restrict__ bias, float* __restrict__ h, int n32) {
    int t = blockIdx.x * blockDim.x + threadIdx.x;
    if (t >= n32) return;
    int node = t >> 5;
    int f = t & 31;
    h[t] = agg[t] * inorm[node] + bias[f];
}

// ---------------- WMMA GEMM kernels ----------------
// One wave computes a 16x32 output tile with V_WMMA_F32_16X16X4_F32.
// A fragment (16x4 f32, 2 VGPRs): lane L, elem v -> m = L%16, k = v + 2*(L/16)
// B fragment (4x16 f32, 2 VGPRs): lane L, elem v -> n = L%16, k = v + 2*(L/16)
// C/D (16x16 f32, 8 VGPRs):       lane L, elem r -> n = L%16, m = r + 8*(L/16)

// Y[m, 0:32] = (X[m, 0:128] * norm[m]) @ W(128x32)
__global__ void gcn_gemm1(const float* __restrict__ X, const float* __restrict__ W,
                          const float* __restrict__ norm, float* __restrict__ Y,
                          int ntiles, int Nn) {
    int lane = threadIdx.x & 31;
    int warp = threadIdx.x >> 5;
    int tile = blockIdx.x * WARPS_PER_BLOCK + warp;
    if (tile >= ntiles) return;               // wave-uniform: EXEC stays all-1s
    int mlane = lane & 15;
    int half  = lane >> 4;
    int m = tile * 16 + mlane;
    int mc = m < Nn ? m : (Nn - 1);
    float sc = norm[mc];
    const float* xrow = X + (long long)mc * IN_FEATS;
    v8f acc0 = {};
    v8f acc1 = {};
    #pragma unroll 4
    for (int kk = 0; kk < IN_FEATS; kk += 4) {
        int ka = kk + 2 * half;
        v2f a;  a.x = xrow[ka] * sc;          a.y = xrow[ka + 1] * sc;
        const float* wr = W + ka * HID;
        v2f b0; b0.x = wr[mlane];             b0.y = wr[HID + mlane];
        v2f b1; b1.x = wr[mlane + 16];        b1.y = wr[HID + mlane + 16];
        acc0 = __builtin_amdgcn_wmma_f32_16x16x4_f32(false, a, false, b0, (short)0, acc0, false, false);
        acc1 = __builtin_amdgcn_wmma_f32_16x16x4_f32(false, a, false, b1, (short)0, acc1, false, false);
    }
    float* yrow = Y + (tile * 16 + 8 * half) * HID + mlane;
    if (tile * 16 + 16 <= Nn) {               // wave-uniform full-tile fast path
        #pragma unroll
        for (int r = 0; r < 8; r++) {
            yrow[r * HID]      = acc0[r];
            yrow[r * HID + 16] = acc1[r];
        }
    } else {                                  // tail tile (only if N % 16 != 0)
        #pragma unroll
        for (int r = 0; r < 8; r++) {
            if (tile * 16 + r + 8 * half < Nn) {
                yrow[r * HID]      = acc0[r];
                yrow[r * HID + 16] = acc1[r];
            }
        }
    }
}

// Y[m, 0:32] = (AG[m, 0:32] * inorm[m]) @ W(32x32) + bias
__global__ void gcn_gemm2(const float* __restrict__ AG, const float* __restrict__ W,
                          const float* __restrict__ inorm, const float* __restrict__ bias,
                          float* __restrict__ Y, int ntiles, int Nn) {
    int lane = threadIdx.x & 31;
    int warp = threadIdx.x >> 5;
    int tile = blockIdx.x * WARPS_PER_BLOCK + warp;
    if (tile >= ntiles) return;
    int mlane = lane & 15;
    int half  = lane >> 4;
    int m = tile * 16 + mlane;
    int mc = m < Nn ? m : (Nn - 1);
    float sc = inorm[mc];
    const float* xrow = AG + mc * HID;
    v8f acc0 = {};
    v8f acc1 = {};
    #pragma unroll
    for (int kk = 0; kk < HID; kk += 4) {
        int ka = kk + 2 * half;
        v2f a;  a.x = xrow[ka] * sc;          a.y = xrow[ka + 1] * sc;
        const float* wr = W + ka * HID;
        v2f b0; b0.x = wr[mlane];             b0.y = wr[HID + mlane];
        v2f b1; b1.x = wr[mlane + 16];        b1.y = wr[HID + mlane + 16];
        acc0 = __builtin_amdgcn_wmma_f32_16x16x4_f32(false, a, false, b0, (short)0, acc0, false, false);
        acc1 = __builtin_amdgcn_wmma_f32_16x16x4_f32(false, a, false, b1, (short)0, acc1, false, false);
    }
    float bb0 = bias[mlane];
    float bb1 = bias[mlane + 16];
    float* yrow = Y + (tile * 16 + 8 * half) * HID + mlane;
    if (tile * 16 + 16 <= Nn) {               // wave-uniform full-tile fast path
        #pragma unroll
        for (int r = 0; r < 8; r++) {
            yrow[r * HID]      = acc0[r] + bb0;
            yrow[r * HID + 16] = acc1[r] + bb1;
        }
    } else {
        #pragma unroll
        for (int r = 0; r < 8; r++) {
            if (tile * 16 + r + 8 * half < Nn) {
                yrow[r * HID]      = acc0[r] + bb0;
                yrow[r * HID + 16] = acc1[r] + bb1;
            }
        }
    }
}

// ---------------- launch ----------------

extern "C" void kernel_launch(void* const* d_in, const int* in_sizes, int n_in,
                              void* d_out, int out_size, void* d_ws, size_t ws_size,
                              hipStream_t stream) {
    const float* in_feat = (const float*)d_in[0];
    const float* W1      = (const float*)d_in[1];
    const float* b1      = (const float*)d_in[2];
    const float* W2      = (const float*)d_in[3];
    const float* b2      = (const float*)d_in[4];
    const int*   src     = (const int*)d_in[5];
    const int*   dst     = (const int*)d_in[6];

    const int N = in_sizes[0] / IN_FEATS;
    const int E = in_sizes[5];

    float* ws    = (float*)d_ws;
    float* onorm = ws;                 // N  (degree accum -> out_norm)
    float* inorm = ws + N;             // N  (degree accum -> in_norm)
    float* h     = ws + 2 * N;         // N*32
    float* agg   = h + N * HID;        // N*32
    float* out   = (float*)d_out;

    const int T = 256;
    const int n32 = N * HID;
    const int ntiles = (N + 15) / 16;
    const int gemm_blocks = (ntiles + WARPS_PER_BLOCK - 1) / WARPS_PER_BLOCK;

    // degrees -> norms
    gcn_fill0<<<(2 * N + T - 1) / T, T, 0, stream>>>(onorm, 2 * N);
    gcn_degree<<<(E + T - 1) / T, T, 0, stream>>>(src, dst, onorm, inorm, E);
    gcn_rsqrt<<<(2 * N + T - 1) / T, T, 0, stream>>>(onorm, 2 * N);

    // conv1: h = (x * out_norm) @ W1  (WMMA), then SpMM, then *in_norm + b1
    gcn_gemm1<<<gemm_blocks, WARPS_PER_BLOCK * 32, 0, stream>>>(in_feat, W1, onorm, h, ntiles, N);
    gcn_fill0<<<(n32 + T - 1) / T, T, 0, stream>>>(agg, n32);
    gcn_spmm<<<((E * 8) + T - 1) / T, T, 0, stream>>>(h, src, dst, onorm, agg, E, 0);
    gcn_post1<<<(n32 + T - 1) / T, T, 0, stream>>>(agg, inorm, b1, h, n32);

    // conv2 x7: SpMM(h * out_norm) -> (agg * in_norm) @ W2 + b2  (WMMA)
    for (int s = 1; s < 8; s++) {
        gcn_fill0<<<(n32 + T - 1) / T, T, 0, stream>>>(agg, n32);
        gcn_spmm<<<((E * 8) + T - 1) / T, T, 0, stream>>>(h, src, dst, onorm, agg, E, 1);
        float* dstp = (s == 7) ? out : h;
        gcn_gemm2<<<gemm_blocks, WARPS_PER_BLOCK * 32, 0, stream>>>(agg, W2, inorm, b2, dstp, ntiles, N);
    }
}